// NeuralTuringMachineMemory_9929964388695
// MI455X (gfx1250) — compile-verified
//
#include <hip/hip_runtime.h>

// Problem dims
#define B_ 4096
#define N_ 128
#define D_ 64
#define H_ 256
#define O_ 128
#define G_ 1024   // 4*H

typedef __attribute__((ext_vector_type(2))) float v2f;
typedef __attribute__((ext_vector_type(4))) float v4f;
typedef __attribute__((ext_vector_type(8))) float v8f;

// fp32 WMMA: D(16x16) = A(16x4) * B(4x16) + C.
// Fragment layout (ISA 7.12.2): lanes 0-15 hold K=k0,k0+1; lanes 16-31 K=k0+2,k0+3.
// C/D: lane half h, VGPR r -> element [M = r + 8*h][N = lane&15].
__device__ __forceinline__ v8f wmma4(v2f a, v2f b, v8f c) {
  return __builtin_amdgcn_wmma_f32_16x16x4_f32(
      /*neg_a=*/false, a, /*neg_b=*/false, b,
      /*c_mod=*/(short)0, c, /*reuse_a=*/false, /*reuse_b=*/false);
}

__device__ __forceinline__ float sigmoidf_(float x) {
  return 1.0f / (1.0f + __expf(-x));
}

#define V8_ZERO {0.f,0.f,0.f,0.f,0.f,0.f,0.f,0.f}

// ---------------------------------------------------------------------------
// Kernel 1: read head. Per 16-row batch tile:
//   S = h_tile @ W_read^T + b_read  (WMMA, 16x128)
//   read_w = softmax(S)             (in LDS)
//   read_vec = read_w @ memory[b]   (B128 streaming; memory stays L2-resident
//                                    for the later update pass)
// ---------------------------------------------------------------------------
__global__ __launch_bounds__(256) void k_read(const float* __restrict__ h,
                                              const float* __restrict__ W_read,
                                              const float* __restrict__ b_read,
                                              const float* __restrict__ memory,
                                              float* __restrict__ read_vec) {
  __shared__ __align__(16) float sA[16 * 260];  // h tile, stride 260 (== 4 mod 64)
  __shared__ __align__(16) float sS[16 * 132];  // scores -> read_w
  const int b0 = blockIdx.x * 16;
  const int tid = threadIdx.x;

  for (int s = tid; s < 16 * 64; s += 256) {    // 64 float4 per 256-wide row
    int r = s >> 6, k = (s & 63) * 4;
    *(v4f*)&sA[r * 260 + k] = *(const v4f*)(h + (size_t)(b0 + r) * H_ + k);
  }
  __syncthreads();

  const int wv = tid >> 5, lane = tid & 31, half = lane >> 4, l16 = lane & 15;
  const int n = wv * 16 + l16;                  // score column 0..127
  const float bias = b_read[n];

  v8f acc = V8_ZERO;
  for (int k0 = 0; k0 < H_; k0 += 4) {
    const int kk = k0 + 2 * half;
    v2f a, b;
    a.x = sA[l16 * 260 + kk];
    a.y = sA[l16 * 260 + kk + 1];
    const float* wp = W_read + (size_t)n * H_ + kk;   // B[k][n] = W_read[n][k]
    b.x = wp[0];
    b.y = wp[1];
    acc = wmma4(a, b, acc);
  }
#pragma unroll
  for (int r = 0; r < 8; ++r)
    sS[(r + 8 * half) * 132 + n] = acc[r] + bias;
  __syncthreads();

  // row softmax over N=128 (one thread per batch row)
  if (tid < 16) {
    float m = -1e30f;
    for (int j = 0; j < N_; ++j) m = fmaxf(m, sS[tid * 132 + j]);
    float s = 0.f;
    for (int j = 0; j < N_; ++j) {
      float e = __expf(sS[tid * 132 + j] - m);
      sS[tid * 132 + j] = e;
      s += e;
    }
    float inv = 1.0f / s;
    for (int j = 0; j < N_; ++j) sS[tid * 132 + j] *= inv;
  }
  __syncthreads();

  // read_vec = read_w @ memory : 16 lane-groups x 16 lanes, float4 over d
  const int d4 = (tid & 15) * 4;                // 0..60
  const int bl = tid >> 4;                      // 0..15
  const float* mp = memory + ((size_t)(b0 + bl) * N_) * D_ + d4;
  float ax = 0.f, ay = 0.f, az = 0.f, aw = 0.f;
  for (int nn = 0; nn < N_; ++nn) {
    const float w = sS[bl * 132 + nn];
    const v4f m = *(const v4f*)(mp + (size_t)nn * D_);
    ax += w * m.x; ay += w * m.y; az += w * m.z; aw += w * m.w;
  }
  v4f r4; r4.x = ax; r4.y = ay; r4.z = az; r4.w = aw;
  *(v4f*)(read_vec + (size_t)(b0 + bl) * D_ + d4) = r4;
}

// ---------------------------------------------------------------------------
// Kernel 2: gates = [x | read_vec | h] @ [W_ih | W_hh]^T + b_ih + b_hh
// K = 448 concat staged in LDS. Grid: (B/16 row tiles) x (8 column blocks of 128).
// ---------------------------------------------------------------------------
__global__ __launch_bounds__(256) void k_gates(const float* __restrict__ x,
                                               const float* __restrict__ read_vec,
                                               const float* __restrict__ h,
                                               const float* __restrict__ W_ih,
                                               const float* __restrict__ W_hh,
                                               const float* __restrict__ b_ih,
                                               const float* __restrict__ b_hh,
                                               float* __restrict__ gates) {
  __shared__ __align__(16) float sA[16 * 452];  // 452 % 64 == 4
  const int b0 = blockIdx.x * 16;
  const int c0 = blockIdx.y * 128;
  const int tid = threadIdx.x;

  for (int s = tid; s < 16 * 112; s += 256) {   // 112 float4 per 448-wide row
    int r = s / 112, k = (s % 112) * 4;         // segment bounds are mult of 4
    v4f v;
    if (k < 128)      v = *(const v4f*)(x + (size_t)(b0 + r) * O_ + k);
    else if (k < 192) v = *(const v4f*)(read_vec + (size_t)(b0 + r) * D_ + (k - 128));
    else              v = *(const v4f*)(h + (size_t)(b0 + r) * H_ + (k - 192));
    *(v4f*)&sA[r * 452 + k] = v;
  }
  __syncthreads();

  const int wv = tid >> 5, lane = tid & 31, half = lane >> 4, l16 = lane & 15;
  const int j = c0 + wv * 16 + l16;             // gate column 0..1023

  v8f acc = V8_ZERO;
  for (int k0 = 0; k0 < 448; k0 += 4) {
    const int kk = k0 + 2 * half;
    v2f a, b;
    a.x = sA[l16 * 452 + kk];
    a.y = sA[l16 * 452 + kk + 1];
    const float* wp = (kk < 192) ? (W_ih + (size_t)j * 192 + kk)
                                 : (W_hh + (size_t)j * 256 + (kk - 192));
    b.x = wp[0];
    b.y = wp[1];
    acc = wmma4(a, b, acc);
  }
  const float bias = b_ih[j] + b_hh[j];
#pragma unroll
  for (int r = 0; r < 8; ++r)
    gates[(size_t)(b0 + r + 8 * half) * G_ + j] = acc[r] + bias;
}

// ---------------------------------------------------------------------------
// Kernel 3: LSTM elementwise (float4) -> h_new, c_new (d_out slices)
// ---------------------------------------------------------------------------
__global__ __launch_bounds__(256) void k_lstm(const float* __restrict__ gates,
                                              const float* __restrict__ c,
                                              float* __restrict__ h_new,
                                              float* __restrict__ c_new) {
  const int t = blockIdx.x * 256 + threadIdx.x;   // 0 .. B*H/4-1
  const int b = t >> 6;
  const int j = (t & 63) * 4;
  const float* g = gates + (size_t)b * G_;
  const v4f gi = *(const v4f*)(g + j);
  const v4f gf = *(const v4f*)(g + H_ + j);
  const v4f gg = *(const v4f*)(g + 2 * H_ + j);
  const v4f go = *(const v4f*)(g + 3 * H_ + j);
  const v4f cv = *(const v4f*)(c + (size_t)b * H_ + j);
  v4f cn, hn;
#pragma unroll
  for (int i = 0; i < 4; ++i) {
    const float iv = sigmoidf_(gi[i]);
    const float fv = sigmoidf_(gf[i]);
    const float gv = tanhf(gg[i]);
    const float ov = sigmoidf_(go[i]);
    const float cc = fv * cv[i] + iv * gv;
    cn[i] = cc;
    hn[i] = ov * tanhf(cc);
  }
  *(v4f*)(c_new + (size_t)b * H_ + j) = cn;
  *(v4f*)(h_new + (size_t)b * H_ + j) = hn;
}

// ---------------------------------------------------------------------------
// Kernel 4a: write/erase/add heads (WMMA 16x256 per tile) + softmax/sigmoid/
// tanh, results to P scratch: P[b][0..127]=write_w | [128..191]=erase |
// [192..255]=add. Decoupled from the streaming update for occupancy.
// ---------------------------------------------------------------------------
__global__ __launch_bounds__(256) void k_whead(const float* __restrict__ h_new,
                                               const float* __restrict__ W_write,
                                               const float* __restrict__ b_write,
                                               const float* __restrict__ W_erase,
                                               const float* __restrict__ b_erase,
                                               const float* __restrict__ W_add,
                                               const float* __restrict__ b_add,
                                               float* __restrict__ P) {
  __shared__ __align__(16) float sH[16 * 260];
  __shared__ __align__(16) float sP[16 * 260];
  const int b0 = blockIdx.x * 16;
  const int tid = threadIdx.x;

  for (int s = tid; s < 16 * 64; s += 256) {
    int r = s >> 6, k = (s & 63) * 4;
    *(v4f*)&sH[r * 260 + k] = *(const v4f*)(h_new + (size_t)(b0 + r) * H_ + k);
  }
  __syncthreads();

  const int wv = tid >> 5, lane = tid & 31, half = lane >> 4, l16 = lane & 15;
  for (int t = wv; t < 16; t += 8) {            // 16 column tiles, 8 waves
    const int jc = t * 16 + l16;                // combined column 0..255
    const float* Wrow;
    float bias;
    if (jc < 128)      { Wrow = W_write + (size_t)jc * H_;         bias = b_write[jc]; }
    else if (jc < 192) { Wrow = W_erase + (size_t)(jc - 128) * H_; bias = b_erase[jc - 128]; }
    else               { Wrow = W_add   + (size_t)(jc - 192) * H_; bias = b_add[jc - 192]; }

    v8f acc = V8_ZERO;
    for (int k0 = 0; k0 < H_; k0 += 4) {
      const int kk = k0 + 2 * half;
      v2f a, b;
      a.x = sH[l16 * 260 + kk];
      a.y = sH[l16 * 260 + kk + 1];
      b.x = Wrow[kk];
      b.y = Wrow[kk + 1];
      acc = wmma4(a, b, acc);
    }
#pragma unroll
    for (int r = 0; r < 8; ++r)
      sP[(r + 8 * half) * 260 + jc] = acc[r] + bias;
  }
  __syncthreads();

  // erase: sigmoid, add: tanh (region disjoint from softmax -> single barrier)
  for (int idx = tid; idx < 16 * 128; idx += 256) {
    int r = idx >> 7, cc = idx & 127;
    float v = sP[r * 260 + 128 + cc];
    sP[r * 260 + 128 + cc] = (cc < 64) ? sigmoidf_(v) : tanhf(v);
  }
  if (tid < 16) {                               // softmax over write columns
    float m = -1e30f;
    for (int j = 0; j < 128; ++j) m = fmaxf(m, sP[tid * 260 + j]);
    float s = 0.f;
    for (int j = 0; j < 128; ++j) {
      float e = __expf(sP[tid * 260 + j] - m);
      sP[tid * 260 + j] = e;
      s += e;
    }
    float inv = 1.0f / s;
    for (int j = 0; j < 128; ++j) sP[tid * 260 + j] *= inv;
  }
  __syncthreads();

  for (int s = tid; s < 16 * 64; s += 256) {    // flush P tile (B128 stores)
    int r = s >> 6, k = (s & 63) * 4;
    *(v4f*)(P + (size_t)(b0 + r) * 256 + k) = *(const v4f*)&sP[r * 260 + k];
  }
}

// ---------------------------------------------------------------------------
// Kernel 4b: mem_new = memory*(1 - ww*erase) + ww*add.
// Pure streaming: 8.4M threads, one float4 each (B128 in / B128 out).
// memory read is last-use (NT) -> should hit L2 from k_read's pass without
// being re-retained; mem_new store is NT (never re-read).
// ---------------------------------------------------------------------------
__global__ __launch_bounds__(256) void k_memupd(const float* __restrict__ memory,
                                                const float* __restrict__ P,
                                                float* __restrict__ mem_new) {
  const size_t idx4 = ((size_t)blockIdx.x * 256 + threadIdx.x) * 4;  // elem offset
  const int d4 = (int)(idx4 & (D_ - 1));        // 0..60
  const size_t bn = idx4 >> 6;                  // b*N + n
  const int nn = (int)(bn & (N_ - 1));
  const size_t b = bn >> 7;
  const float* Pb = P + b * 256;
  const float w = Pb[nn];
  const v4f e = *(const v4f*)(Pb + 128 + d4);
  const v4f a = *(const v4f*)(Pb + 192 + d4);
  const v4f m = __builtin_nontemporal_load((const v4f*)(memory + idx4));
  v4f o;
  o.x = m.x * (1.0f - w * e.x) + w * a.x;
  o.y = m.y * (1.0f - w * e.y) + w * a.y;
  o.z = m.z * (1.0f - w * e.z) + w * a.z;
  o.w = m.w * (1.0f - w * e.w) + w * a.w;
  __builtin_nontemporal_store(o, (v4f*)(mem_new + idx4));
}

// ---------------------------------------------------------------------------
// Kernel 5: out = [h_new | read_vec] @ W_out^T + b_out   (K=320, WMMA)
// ---------------------------------------------------------------------------
__global__ __launch_bounds__(256) void k_out(const float* __restrict__ h_new,
                                             const float* __restrict__ read_vec,
                                             const float* __restrict__ W_out,
                                             const float* __restrict__ b_out,
                                             float* __restrict__ out) {
  __shared__ __align__(16) float sA[16 * 324];  // 324 % 64 == 4
  const int b0 = blockIdx.x * 16;
  const int tid = threadIdx.x;

  for (int s = tid; s < 16 * 80; s += 256) {    // 80 float4 per 320-wide row
    int r = s / 80, k = (s % 80) * 4;
    v4f v;
    if (k < 256) v = *(const v4f*)(h_new + (size_t)(b0 + r) * H_ + k);
    else         v = *(const v4f*)(read_vec + (size_t)(b0 + r) * D_ + (k - 256));
    *(v4f*)&sA[r * 324 + k] = v;
  }
  __syncthreads();

  const int wv = tid >> 5, lane = tid & 31, half = lane >> 4, l16 = lane & 15;
  const int j = wv * 16 + l16;                  // output column 0..127

  v8f acc = V8_ZERO;
  for (int k0 = 0; k0 < 320; k0 += 4) {
    const int kk = k0 + 2 * half;
    v2f a, b;
    a.x = sA[l16 * 324 + kk];
    a.y = sA[l16 * 324 + kk + 1];
    const float* wp = W_out + (size_t)j * 320 + kk;
    b.x = wp[0];
    b.y = wp[1];
    acc = wmma4(a, b, acc);
  }
  const float bias = b_out[j];
#pragma unroll
  for (int r = 0; r < 8; ++r)
    out[(size_t)(b0 + r + 8 * half) * O_ + j] = acc[r] + bias;
}

// ---------------------------------------------------------------------------
extern "C" void kernel_launch(void* const* d_in, const int* in_sizes, int n_in,
                              void* d_out, int out_size, void* d_ws, size_t ws_size,
                              hipStream_t stream) {
  const float* x       = (const float*)d_in[0];
  const float* memory  = (const float*)d_in[1];
  const float* h       = (const float*)d_in[2];
  const float* c       = (const float*)d_in[3];
  const float* W_ih    = (const float*)d_in[4];
  const float* W_hh    = (const float*)d_in[5];
  const float* b_ih    = (const float*)d_in[6];
  const float* b_hh    = (const float*)d_in[7];
  const float* W_read  = (const float*)d_in[8];
  const float* b_read  = (const float*)d_in[9];
  const float* W_write = (const float*)d_in[10];
  const float* b_write = (const float*)d_in[11];
  const float* W_erase = (const float*)d_in[12];
  const float* b_erase = (const float*)d_in[13];
  const float* W_add   = (const float*)d_in[14];
  const float* b_add   = (const float*)d_in[15];
  const float* W_out   = (const float*)d_in[16];
  const float* b_out   = (const float*)d_in[17];

  // d_out: out (B*O) | mem_new (B*N*D) | h_new (B*H) | c_new (B*H)
  float* out     = (float*)d_out;
  float* mem_new = out + (size_t)B_ * O_;
  float* h_new   = mem_new + (size_t)B_ * N_ * D_;
  float* c_new   = h_new + (size_t)B_ * H_;

  // workspace: read_vec (B*D) | gates (B*4H) | P (B*256)  (~22 MB)
  float* read_vec = (float*)d_ws;
  float* gates    = read_vec + (size_t)B_ * D_;
  float* P        = gates + (size_t)B_ * G_;

  k_read<<<B_ / 16, 256, 0, stream>>>(h, W_read, b_read, memory, read_vec);
  k_gates<<<dim3(B_ / 16, 8), 256, 0, stream>>>(x, read_vec, h, W_ih, W_hh,
                                                b_ih, b_hh, gates);
  k_lstm<<<(B_ * H_) / (256 * 4), 256, 0, stream>>>(gates, c, h_new, c_new);
  k_whead<<<B_ / 16, 256, 0, stream>>>(h_new, W_write, b_write, W_erase, b_erase,
                                       W_add, b_add, P);
  k_memupd<<<(B_ * N_ * D_) / (256 * 4), 256, 0, stream>>>(memory, P, mem_new);
  k_out<<<B_ / 16, 256, 0, stream>>>(h_new, read_vec, W_out, b_out, out);
}